// Encoder_18897856102728
// MI455X (gfx1250) — compile-verified
//
#include <hip/hip_runtime.h>
#include <hip/hip_bf16.h>

typedef float v2f __attribute__((ext_vector_type(2)));
typedef float v8f __attribute__((ext_vector_type(8)));

// ---------------------------------------------------------------------------
// WMMA fp32 GEMM: C[M,N] = A[M,K] @ W[K,N], row-major, N,K multiples of 16/4.
// One wave (32 lanes) computes one 16x16 tile via V_WMMA_F32_16X16X4_F32.
// blockDim = (32, N/16): waves share a 16-row A strip staged in LDS.
// Fragment layouts per CDNA5 ISA 7.12.2:
//   A 16x4 : lanes 0-15 hold (M=lane, K=k..k+1), lanes 16-31 (M=lane-16, K=k+2..k+3)
//   B 4x16 : lanes 0-15 hold (N=lane, K=k..k+1), lanes 16-31 (N=lane-16, K=k+2..k+3)
//   D 16x16: VGPR r -> M = r + 8*(lane/16), N = lane%16
// ---------------------------------------------------------------------------
__global__ void wmma_gemm_f32(const float* __restrict__ A,
                              const float* __restrict__ W,
                              float* __restrict__ C,
                              int M, int N, int K) {
    extern __shared__ float ldsA[];                 // [16][K]
    const int rowBase = blockIdx.x * 16;
    const int lane    = threadIdx.x;                // 0..31
    const int colBase = threadIdx.y * 16;
    const int tid     = threadIdx.y * 32 + lane;
    const int nthr    = blockDim.x * blockDim.y;

    // Stage the 16-row A strip (contiguous in global since lda == K).
    const int rows  = (M - rowBase) < 16 ? (M - rowBase) : 16;
    const int strip = rows * K;
    for (int i = tid; i < strip; i += nthr)
        ldsA[i] = A[(size_t)rowBase * K + i];
    __syncthreads();

    const int half = lane >> 4;                     // 0 | 1
    const int mrow = lane & 15;                     // M (for A) / N (for B,D) index
    const int ncol = colBase + mrow;

    v8f acc = {};
    for (int k = 0; k < K; k += 4) {
        const int ka = k + half * 2;
        // A fragment: two consecutive K values -> 8B LDS load
        v2f a = *(const v2f*)&ldsA[mrow * K + ka];
        // B fragment: W[ka][ncol], W[ka+1][ncol]
        v2f b;
        b.x = W[(size_t)ka * N + ncol];
        b.y = W[(size_t)(ka + 1) * N + ncol];
        acc = __builtin_amdgcn_wmma_f32_16x16x4_f32(
            /*neg_a=*/false, a, /*neg_b=*/false, b,
            /*c_mod=*/(short)0, acc, /*reuse_a=*/false, /*reuse_b=*/false);
    }

    const int outRow0 = rowBase + half * 8;
#pragma unroll
    for (int r = 0; r < 8; ++r) {
        const int orow = outRow0 + r;
        if (orow < M) C[(size_t)orow * N + ncol] = acc[r];
    }
}

// ---------------------------------------------------------------------------
// Degree / normalization
// ---------------------------------------------------------------------------
__global__ void fzero_kernel(float* __restrict__ p, size_t n) {
    size_t i = (size_t)blockIdx.x * blockDim.x + threadIdx.x;
    if (i < n) p[i] = 0.0f;
}

__global__ void degree_kernel(const int* __restrict__ dst, float* __restrict__ deg, int E) {
    int e = blockIdx.x * blockDim.x + threadIdx.x;
    if (e < E) atomicAdd(&deg[dst[e]], 1.0f);
}

// dinv[i] = rsqrt(deg_edges[i] + 1)   (+1 = self loop; always > 0)
__global__ void dinv_kernel(float* __restrict__ deg, int N) {
    int i = blockIdx.x * blockDim.x + threadIdx.x;
    if (i < N) deg[i] = rsqrtf(deg[i] + 1.0f);
}

// ---------------------------------------------------------------------------
// Self-loop initialization: agg[i][c] = h[i][c] * dinv[i]^2  (ch = 1<<shift)
// ---------------------------------------------------------------------------
__global__ void selfloop_init(const float* __restrict__ h,
                              const float* __restrict__ dinv,
                              float* __restrict__ agg, size_t total, int shift) {
    size_t i = (size_t)blockIdx.x * blockDim.x + threadIdx.x;
    if (i < total) {
        float d = dinv[i >> shift];
        agg[i] = h[i] * (d * d);
    }
}

// ---------------------------------------------------------------------------
// Edge scatter, layer 1 (128 channels): one wave per edge, 4 ch / lane.
// ---------------------------------------------------------------------------
__global__ void scatter_l1(const int* __restrict__ src, const int* __restrict__ dst,
                           const float* __restrict__ dinv, const float* __restrict__ h,
                           float* __restrict__ agg, int E) {
    int wave = (int)((blockIdx.x * (size_t)blockDim.x + threadIdx.x) >> 5);
    int lane = threadIdx.x & 31;
    if (wave >= E) return;
    int s = src[wave], d = dst[wave];
    float norm = dinv[s] * dinv[d];
    const float4 v = ((const float4*)(h + (size_t)s * 128))[lane];
    float* out = agg + (size_t)d * 128 + lane * 4;
    atomicAdd(out + 0, v.x * norm);
    atomicAdd(out + 1, v.y * norm);
    atomicAdd(out + 2, v.z * norm);
    atomicAdd(out + 3, v.w * norm);
}

// ---------------------------------------------------------------------------
// Edge scatter, layer 2 fused mu+logstd (64+64 channels): 2+2 ch / lane.
// ---------------------------------------------------------------------------
__global__ void scatter_l2(const int* __restrict__ src, const int* __restrict__ dst,
                           const float* __restrict__ dinv,
                           const float* __restrict__ hmu, const float* __restrict__ hls,
                           float* __restrict__ mu, float* __restrict__ ls, int E) {
    int wave = (int)((blockIdx.x * (size_t)blockDim.x + threadIdx.x) >> 5);
    int lane = threadIdx.x & 31;
    if (wave >= E) return;
    int s = src[wave], d = dst[wave];
    float norm = dinv[s] * dinv[d];
    const float2 vm = ((const float2*)(hmu + (size_t)s * 64))[lane];
    const float2 vl = ((const float2*)(hls + (size_t)s * 64))[lane];
    float* om = mu + (size_t)d * 64 + lane * 2;
    float* ol = ls + (size_t)d * 64 + lane * 2;
    atomicAdd(om + 0, vm.x * norm);
    atomicAdd(om + 1, vm.y * norm);
    atomicAdd(ol + 0, vl.x * norm);
    atomicAdd(ol + 1, vl.y * norm);
}

// ---------------------------------------------------------------------------
// Epilogues
// ---------------------------------------------------------------------------
__global__ void bias_relu_kernel(float* __restrict__ a, const float* __restrict__ b,
                                 size_t total, int mask) {
    size_t i = (size_t)blockIdx.x * blockDim.x + threadIdx.x;
    if (i < total) a[i] = fmaxf(a[i] + b[i & mask], 0.0f);
}

__global__ void bias_add_kernel(float* __restrict__ a, const float* __restrict__ b,
                                size_t total, int mask) {
    size_t i = (size_t)blockIdx.x * blockDim.x + threadIdx.x;
    if (i < total) a[i] += b[i & mask];
}

// ---------------------------------------------------------------------------
extern "C" void kernel_launch(void* const* d_in, const int* in_sizes, int n_in,
                              void* d_out, int out_size, void* d_ws, size_t ws_size,
                              hipStream_t stream) {
    const float* x   = (const float*)d_in[0];
    const int*   ei  = (const int*)d_in[1];     // [2][E] : row0 = src, row1 = dst
    const float* W1  = (const float*)d_in[2];
    const float* b1  = (const float*)d_in[3];
    const float* Wmu = (const float*)d_in[4];
    const float* bmu = (const float*)d_in[5];
    const float* Wls = (const float*)d_in[6];
    const float* bls = (const float*)d_in[7];

    const int IN = 256, HID = 128, OUT = 64;
    const int N  = in_sizes[0] / IN;            // 100000
    const int E  = in_sizes[1] / 2;             // 1600000
    const int* srcIdx = ei;
    const int* dstIdx = ei + E;

    // workspace layout (fp32)
    float* dinv = (float*)d_ws;                                  // N
    float* xW1  = dinv + ((N + 255) & ~255);                     // N*HID
    float* h    = xW1 + (size_t)N * HID;                         // N*HID (agg -> relu'd h)
    float* hmu  = h + (size_t)N * HID;                           // N*OUT
    float* hls  = hmu + (size_t)N * OUT;                         // N*OUT

    float* muOut = (float*)d_out;                                // N*OUT
    float* lsOut = muOut + (size_t)N * OUT;                      // N*OUT

    const int TPB = 256;
    // --- degrees / normalization ------------------------------------------
    fzero_kernel<<<(N + TPB - 1) / TPB, TPB, 0, stream>>>(dinv, (size_t)N);
    degree_kernel<<<(E + TPB - 1) / TPB, TPB, 0, stream>>>(dstIdx, dinv, E);
    dinv_kernel<<<(N + TPB - 1) / TPB, TPB, 0, stream>>>(dinv, N);

    // --- layer 1: xW1 = x @ W1 (WMMA), self-loop init, scatter, bias+relu --
    {
        dim3 blk(32, HID / 16);                   // 8 waves
        dim3 grd((N + 15) / 16);
        wmma_gemm_f32<<<grd, blk, 16 * IN * sizeof(float), stream>>>(x, W1, xW1, N, HID, IN);
    }
    size_t tot1 = (size_t)N * HID;
    selfloop_init<<<(tot1 + TPB - 1) / TPB, TPB, 0, stream>>>(xW1, dinv, h, tot1, 7);
    scatter_l1<<<(E + 7) / 8, TPB, 0, stream>>>(srcIdx, dstIdx, dinv, xW1, h, E);
    bias_relu_kernel<<<(tot1 + TPB - 1) / TPB, TPB, 0, stream>>>(h, b1, tot1, HID - 1);

    // --- layer 2: hmu = h @ Wmu, hls = h @ Wls (WMMA) ----------------------
    {
        dim3 blk(32, OUT / 16);                   // 4 waves
        dim3 grd((N + 15) / 16);
        wmma_gemm_f32<<<grd, blk, 16 * HID * sizeof(float), stream>>>(h, Wmu, hmu, N, OUT, HID);
        wmma_gemm_f32<<<grd, blk, 16 * HID * sizeof(float), stream>>>(h, Wls, hls, N, OUT, HID);
    }
    size_t tot2 = (size_t)N * OUT;
    selfloop_init<<<(tot2 + TPB - 1) / TPB, TPB, 0, stream>>>(hmu, dinv, muOut, tot2, 6);
    selfloop_init<<<(tot2 + TPB - 1) / TPB, TPB, 0, stream>>>(hls, dinv, lsOut, tot2, 6);
    scatter_l2<<<(E + 7) / 8, TPB, 0, stream>>>(srcIdx, dstIdx, dinv, hmu, hls, muOut, lsOut, E);
    bias_add_kernel<<<(tot2 + TPB - 1) / TPB, TPB, 0, stream>>>(muOut, bmu, tot2, OUT - 1);
    bias_add_kernel<<<(tot2 + TPB - 1) / TPB, TPB, 0, stream>>>(lsOut, bls, tot2, OUT - 1);
}